// PredictHandler_84387517432126
// MI455X (gfx1250) — compile-verified
//
#include <hip/hip_runtime.h>
#include <math.h>
#include <stdint.h>

#define KSEL 1024
#define THR_CONF 0.5f
#define THR_NMS  0.3f
#define VAR0 0.1f
#define VAR1 0.2f

typedef unsigned int u32;
typedef u32 u32x4 __attribute__((ext_vector_type(4)));
typedef int  i32x8 __attribute__((ext_vector_type(8)));
typedef int  i32x4 __attribute__((ext_vector_type(4)));

// ---------------------------------------------------------------------------
// TDM: 1-D DMA of `tile=1024` f32 elements global -> LDS. tensor_dim0 = rem so
// out-of-range tail reads return zero (ISA 08 §8.2). Descriptor packing per
// D# group0/group1 bit layout (ISA 08 §8.3/8.4).
// ---------------------------------------------------------------------------
__device__ __forceinline__ void tdm_load_1d_f32(const float* gsrc, void* lds_dst, int rem) {
  unsigned long long ga = (unsigned long long)(uintptr_t)gsrc;
  u32 lds_off = (u32)(uintptr_t)lds_dst;          // flat LDS addr: low 32 bits = LDS offset
  u32x4 g0 = {
    1u,                                            // count=1, no gather
    lds_off,                                       // lds_addr
    (u32)(ga & 0xFFFFFFFFull),                     // global_addr[31:0]
    (u32)((ga >> 32) & 0x01FFFFFFull) | (2u << 30) // global_addr[56:32] | type=2
  };
  u32 td0 = (u32)(rem > 0 ? rem : 0);
  i32x8 g1 = {
    (int)(2u << 16),                               // data_size = 4 bytes
    (int)((td0 & 0xFFFFu) << 16),                  // tensor_dim0[15:0]
    (int)(((td0 >> 16) & 0xFFFFu) | (1u << 16)),   // tensor_dim0[31:16] | tensor_dim1=1
    (int)((u32)KSEL << 16),                        // tile_dim0 = 1024
    1,                                             // tile_dim1 = 1
    KSEL,                                          // tensor_dim0_stride
    0, 0
  };
  i32x4 z4 = {0, 0, 0, 0};
#if defined(__clang_major__) && __clang_major__ >= 23
  i32x8 z8 = {0, 0, 0, 0, 0, 0, 0, 0};
  __builtin_amdgcn_tensor_load_to_lds(g0, g1, z4, z4, z8, 0);
#else
  __builtin_amdgcn_tensor_load_to_lds(g0, g1, z4, z4, 0);
#endif
}

// strict total order: higher score first, ties -> lower original index first
__device__ __forceinline__ bool pair_gt(float va, int ia, float vb, int ib) {
  return (va > vb) || (va == vb && ia < ib);
}

__device__ __forceinline__ void bitonic_sort_desc(float* v, int* ix, int t) {
  for (int k = 2; k <= KSEL; k <<= 1) {
    for (int j = k >> 1; j > 0; j >>= 1) {
      __syncthreads();
      int p = t ^ j;
      if (p > t) {
        bool desc = ((t & k) == 0);
        float va = v[t]; int ia = ix[t];
        float vb = v[p]; int ib = ix[p];
        bool sw = desc ? pair_gt(vb, ib, va, ia) : pair_gt(va, ia, vb, ib);
        if (sw) { v[t] = vb; ix[t] = ib; v[p] = va; ix[p] = ia; }
      }
    }
  }
  __syncthreads();
}

__device__ __forceinline__ void bitonic_merge_desc(float* v, int* ix, int t) {
  for (int j = KSEL >> 1; j > 0; j >>= 1) {
    __syncthreads();
    int p = t ^ j;
    if (p > t) {
      float va = v[t]; int ia = ix[t];
      float vb = v[p]; int ib = ix[p];
      if (pair_gt(vb, ib, va, ia)) { v[t] = vb; ix[t] = ib; v[p] = va; ix[p] = ia; }
    }
  }
  __syncthreads();
}

// ---------------------------------------------------------------------------
// Kernel 1: per-image exact sorted top-1024 of masked sigmoid scores.
// One 1024-thread workgroup per image. Confidence stream is DMA'd into LDS by
// the Tensor Data Mover, double buffered: issue chunk c+1, s_wait_tensorcnt(1)
// for chunk c, then bitonic sort/merge while the next DMA is in flight.
// ---------------------------------------------------------------------------
__global__ void __launch_bounds__(1024, 1)
rf_topk_kernel(const float* __restrict__ p_conf,
               float* __restrict__ wvals, int* __restrict__ widx, int N) {
  __shared__ float stage[2][KSEL];
  __shared__ float cv[KSEL];  __shared__ int ci[KSEL];
  __shared__ float tv[KSEL];  __shared__ int ti[KSEL];

  const int b = blockIdx.x;
  const int t = threadIdx.x;
  const float* conf = p_conf + (size_t)b * (size_t)N;
  const int nchunks = (N + KSEL - 1) / KSEL;

  if (t < 32) {                       // wave 0 owns the TDM queue
    tdm_load_1d_f32(conf, &stage[0][0], N);
  }

  for (int c = 0; c < nchunks; ++c) {
    if (t < 32) {
      if (c + 1 < nchunks) {
        tdm_load_1d_f32(conf + (size_t)(c + 1) * KSEL, &stage[(c + 1) & 1][0],
                        N - (c + 1) * KSEL);
        __builtin_amdgcn_s_wait_tensorcnt(1);   // chunk c done, c+1 in flight
      } else {
        __builtin_amdgcn_s_wait_tensorcnt(0);
      }
    }
    __syncthreads();

    int g = c * KSEL + t;
    bool valid = g < N;
    float logit = stage[c & 1][t];
    float s = 1.0f / (1.0f + expf(-logit));
    float mv = (s >= THR_CONF) ? s : -1.0f;     // conf mask, as in reference
    cv[t] = valid ? mv : -1.0f;
    ci[t] = valid ? g : 0x7FFFFFFF;             // sentinel loses all ties

    bitonic_sort_desc(cv, ci, t);               // chunk sorted descending

    if (c == 0) {
      tv[t] = cv[t]; ti[t] = ci[t];
      __syncthreads();
    } else {
      // top-k(A ∪ B) = sort( max(A[i], B[K-1-i]) ) — result is bitonic
      float va = tv[t]; int ia = ti[t];
      float vb = cv[KSEL - 1 - t]; int ib = ci[KSEL - 1 - t];
      if (pair_gt(vb, ib, va, ia)) { tv[t] = vb; ti[t] = ib; }
      bitonic_merge_desc(tv, ti, t);
    }
  }

  wvals[(size_t)b * KSEL + t] = tv[t];
  widx [(size_t)b * KSEL + t] = ti[t];
}

// ---------------------------------------------------------------------------
// Kernel 2: decode only the selected 1024 candidates, then column-parallel
// greedy NMS: thread j owns column j; row i broadcast from LDS; one barrier
// per row. Suppressed/low-score columns skip the IoU math.
// ---------------------------------------------------------------------------
__global__ void __launch_bounds__(1024, 1)
rf_nms_kernel(const float* __restrict__ p_loc, const float* __restrict__ p_landms,
              const float* __restrict__ anchors,
              const float* __restrict__ wvals, const int* __restrict__ widx,
              float* __restrict__ out, int N) {
  __shared__ float sl[KSEL], st_[KSEL], sr[KSEL], sb[KSEL];
  __shared__ float ss[KSEL], sa[KSEL];
  __shared__ unsigned int supp[KSEL];

  const int b = blockIdx.x;
  const int j = threadIdx.x;

  float v = wvals[(size_t)b * KSEL + j];
  int id = widx[(size_t)b * KSEL + j];
  if (id < 0 || id >= N) id = 0;                 // memory-safe for filler rows

  const float* a  = anchors  + (size_t)id * 4;
  const float* lc = p_loc    + ((size_t)b * N + id) * 4;
  const float* lm = p_landms + ((size_t)b * N + id) * 10;

  float acx = a[0], acy = a[1], aw = a[2], ah = a[3];
  float cx = acx + lc[0] * VAR0 * aw;
  float cy = acy + lc[1] * VAR0 * ah;
  float bw = aw * expf(lc[2] * VAR1);
  float bh = ah * expf(lc[3] * VAR1);
  float l = cx - bw * 0.5f, tt = cy - bh * 0.5f;
  float r = cx + bw * 0.5f, bo = cy + bh * 0.5f;
  float area = fmaxf(r - l, 0.0f) * fmaxf(bo - tt, 0.0f);

  float kp[10];
#pragma unroll
  for (int p = 0; p < 5; ++p) {
    kp[2 * p + 0] = acx + lm[2 * p + 0] * VAR0 * aw;
    kp[2 * p + 1] = acy + lm[2 * p + 1] * VAR0 * ah;
  }

  sl[j] = l; st_[j] = tt; sr[j] = r; sb[j] = bo;
  ss[j] = v; sa[j] = area; supp[j] = 0u;
  __syncthreads();

  bool alive = (v >= THR_CONF);
  bool suppd = false;
  for (int i = 0; i < KSEL; ++i) {
    bool keep_i = (ss[i] >= THR_CONF) && (supp[i] == 0u);
    if (keep_i && (j > i) && alive && !suppd) {
      float il = fmaxf(l, sl[i]),  it = fmaxf(tt, st_[i]);
      float ir = fminf(r, sr[i]),  ib = fminf(bo, sb[i]);
      float w = fmaxf(ir - il, 0.0f), h = fmaxf(ib - it, 0.0f);
      float inter = w * h;
      float iou = inter / (area + sa[i] - inter + 1e-9f);
      if (iou > THR_NMS) { suppd = true; supp[j] = 1u; }
    }
    __syncthreads();
  }

  float m = (alive && !suppd) ? 1.0f : 0.0f;
  size_t base = ((size_t)b * KSEL + j) * 15;
  out[base + 0] = l  * m;
  out[base + 1] = tt * m;
  out[base + 2] = r  * m;
  out[base + 3] = bo * m;
#pragma unroll
  for (int p = 0; p < 10; ++p) out[base + 4 + p] = kp[p] * m;
  out[base + 14] = v * m;
}

extern "C" void kernel_launch(void* const* d_in, const int* in_sizes, int n_in,
                              void* d_out, int out_size, void* d_ws, size_t ws_size,
                              hipStream_t stream) {
  (void)n_in; (void)out_size; (void)ws_size;
  const float* p_loc    = (const float*)d_in[0];
  const float* p_conf   = (const float*)d_in[1];
  const float* p_landms = (const float*)d_in[2];
  const float* anchors  = (const float*)d_in[3];

  const int N = in_sizes[3] / 4;        // anchors: (N,4)
  const int B = in_sizes[1] / N;        // p_conf:  (B,N,1)

  float* wvals = (float*)d_ws;
  int*   widx  = (int*)((char*)d_ws + (size_t)B * KSEL * sizeof(float));

  rf_topk_kernel<<<B, KSEL, 0, stream>>>(p_conf, wvals, widx, N);
  rf_nms_kernel <<<B, KSEL, 0, stream>>>(p_loc, p_landms, anchors, wvals, widx,
                                         (float*)d_out, N);
}